// LIFNeuron_30116310680258
// MI455X (gfx1250) — compile-verified
//
#include <hip/hip_runtime.h>

// LIF neuron scan: v = tau*v + x_t; spike = (v >= thr); v = 0 on spike.
// Memory-bound streaming scan (256 MiB total @ 23.3 TB/s => ~11us floor).
// Strategy: one f32 membrane state per thread (65536 threads = 2048 wave32s),
// sequential loop over T with 4x unroll for memory-level parallelism,
// non-temporal loads/stores (read-once/write-once streams), and
// global_prefetch_b8 (gfx1250) 8 timesteps (2 MiB) ahead to warm L2.

namespace {
constexpr int   kT      = 512;
constexpr int   kB      = 64;
constexpr int   kD      = 1024;
constexpr int   kBD     = kB * kD;   // 65536 independent LIF lanes
constexpr float kTau    = 0.5f;
constexpr float kThr    = 1.0f;
constexpr float kVReset = 0.0f;
constexpr int   kBlock  = 256;       // 8 wave32s per workgroup
constexpr int   kUnroll = 4;         // independent loads in flight per thread
constexpr int   kPfDist = 8;         // prefetch distance in timesteps
}  // namespace

__global__ __launch_bounds__(kBlock) void lif_scan_kernel(
    const float* __restrict__ x,   // [T, B, D]
    const float* __restrict__ v0,  // [D], broadcast over batch
    float* __restrict__ out) {     // [T, B, D] spikes
  const int lane = blockIdx.x * kBlock + threadIdx.x;  // 0 .. kBD-1
  const float* __restrict__ xp = x + lane;
  float* __restrict__ op = out + lane;

  // Initial membrane potential (v0 broadcasts over batch).
  float v = v0[lane & (kD - 1)];

  for (int t = 0; t < kT; t += kUnroll) {
    // Warm L2 a couple MiB ahead in the x stream (global_prefetch_b8).
    if (t + kPfDist < kT) {
      __builtin_prefetch(xp + (size_t)(t + kPfDist) * kBD, 0, 1);
    }

    // Issue kUnroll independent non-temporal loads (clause of global_load_b32)
    // before the dependent scan updates, to hide HBM latency.
    float xin[kUnroll];
#pragma unroll
    for (int u = 0; u < kUnroll; ++u) {
      xin[u] = __builtin_nontemporal_load(xp + (size_t)(t + u) * kBD);
    }

    // Sequential LIF updates (true data dependence through v).
#pragma unroll
    for (int u = 0; u < kUnroll; ++u) {
      v = kTau * v + xin[u];
      const bool fire = (v >= kThr);
      __builtin_nontemporal_store(fire ? 1.0f : 0.0f,
                                  op + (size_t)(t + u) * kBD);
      v = fire ? kVReset : v;
    }
  }
}

extern "C" void kernel_launch(void* const* d_in, const int* /*in_sizes*/,
                              int /*n_in*/, void* d_out, int /*out_size*/,
                              void* /*d_ws*/, size_t /*ws_size*/,
                              hipStream_t stream) {
  const float* x  = (const float*)d_in[0];
  const float* v0 = (const float*)d_in[1];
  float* out      = (float*)d_out;

  static_assert(kBD % kBlock == 0, "grid must tile lanes exactly");
  lif_scan_kernel<<<kBD / kBlock, kBlock, 0, stream>>>(x, v0, out);
}